// SiGAT_23038204575879
// MI455X (gfx1250) — compile-verified
//
#include <hip/hip_runtime.h>
#include <hip/hip_bf16.h>
#include <math.h>

typedef __attribute__((ext_vector_type(2))) float v2f;
typedef __attribute__((ext_vector_type(8))) float v8f;
typedef __attribute__((ext_vector_type(4))) int   v4i;

#define NEG_SLOPE 0.2f
#define EPS_F 1e-16f

#if defined(__gfx1250__) && __has_builtin(__builtin_amdgcn_global_load_async_to_lds_b128)
#define HAVE_ASYNC_LDS 1
#else
#define HAVE_ASYNC_LDS 0
#endif

// ---------------------------------------------------------------------------
// gemm64: D[M,64] = op(A)[M,64] @ B[64,64]  (+Cin tile) (+bias[col]) (tanh out)
// V_WMMA_F32_16X16X4_F32; block = 256 threads = 8 waves -> 32x64 output tile.
// B panel (16 KB) staged into LDS with GLOBAL_LOAD_ASYNC_TO_LDS_B128
// (ASYNCcnt) when available; flags are template params so the hot
// instantiation has a branch-free inner loop.
// ---------------------------------------------------------------------------
template <bool TANHA, bool HASC, bool HASBIAS, bool TANHOUT>
__global__ __launch_bounds__(256) void gemm64_kernel(
    const float* __restrict__ A, const float* __restrict__ B,
    const float* __restrict__ Cin, const float* __restrict__ bias,
    float* __restrict__ D, int M)
{
    __shared__ float Bs[64 * 64];   // row-major copy of B
    const int t = threadIdx.x;

#if HAVE_ASYNC_LDS
    #pragma unroll
    for (int i = 0; i < 4; ++i) {
        int off = (i * 256 + t) * 4;            // 16-byte granules, 16 KB total
        __builtin_amdgcn_global_load_async_to_lds_b128(
            (__attribute__((address_space(1))) v4i*)(B + off),
            (__attribute__((address_space(3))) v4i*)(&Bs[off]),
            0, 0);
    }
#if __has_builtin(__builtin_amdgcn_s_wait_asynccnt)
    __builtin_amdgcn_s_wait_asynccnt(0);
#else
    asm volatile("s_wait_asynccnt 0x0" ::: "memory");
#endif
#else
    #pragma unroll
    for (int i = 0; i < 4; ++i) {
        int off = (i * 256 + t) * 4;
        *(float4*)&Bs[off] = *(const float4*)&B[off];
    }
#endif
    __syncthreads();

    const int wave = t >> 5;                // 0..7
    const int lane = t & 31;
    const int row0 = blockIdx.x * 32 + (wave >> 2) * 16;
    const int col0 = (wave & 3) * 16;

    // A operand (ISA 7.12.2, 32-bit A 16x4):
    //   lanes 0-15: row=lane, K={0,1}; lanes 16-31: row=lane-16, K={2,3}
    const int arow  = row0 + (lane & 15);
    const int kb    = (lane >> 4) * 2;      // 0 or 2
    const int arowc = (arow < M) ? arow : (M - 1);
    const int ncol  = col0 + (lane & 15);   // B column for this lane
    // C/D: col = lane&15, rows = row0 + 8*(lane>>4) + j
    const int dcol  = col0 + (lane & 15);
    const int drow0 = row0 + 8 * (lane >> 4);

    v8f c = {};
    if (HASC) {
        #pragma unroll
        for (int j = 0; j < 8; ++j) {
            int r = drow0 + j;
            c[j] = (r < M) ? Cin[(size_t)r * 64 + dcol] : 0.0f;
        }
    }

    const float* ap = A + (size_t)arowc * 64 + kb;
    #pragma unroll
    for (int k0 = 0; k0 < 64; k0 += 4) {
        v2f a; a.x = ap[k0]; a.y = ap[k0 + 1];
        if (TANHA) { a.x = tanhf(a.x); a.y = tanhf(a.y); }
        v2f b;
        b.x = Bs[(k0 + kb) * 64 + ncol];
        b.y = Bs[(k0 + kb + 1) * 64 + ncol];
        // (neg_a, A, neg_b, B, c_mod, C, reuse_a, reuse_b)
        c = __builtin_amdgcn_wmma_f32_16x16x4_f32(false, a, false, b,
                                                  (short)0, c, false, false);
    }

    #pragma unroll
    for (int j = 0; j < 8; ++j) {
        int r = drow0 + j;
        if (r < M) {
            float v = c[j];
            if (HASBIAS) v += bias[dcol];
            if (TANHOUT) v = tanhf(v);
            D[(size_t)r * 64 + dcol] = v;
        }
    }
}

// ---------------------------------------------------------------------------
// Per-node prep: attention logits via wave32 shuffle reduction; init
// segment-max=-inf, denom=0, out row = gat_bias.  One wave per node.
// ---------------------------------------------------------------------------
__global__ __launch_bounds__(256) void node_prep_kernel(
    const float* __restrict__ h, const float* __restrict__ a_src,
    const float* __restrict__ a_dst, const float* __restrict__ bias,
    float* __restrict__ ls, float* __restrict__ ld_,
    float* __restrict__ mx, float* __restrict__ denom,
    float* __restrict__ outp, int N)
{
    int node = blockIdx.x * 8 + (threadIdx.x >> 5);
    int lane = threadIdx.x & 31;
    if (node >= N) return;
    size_t base = (size_t)node * 64;
    float h0 = h[base + lane];
    float h1 = h[base + 32 + lane];
    float s = h0 * a_src[lane] + h1 * a_src[32 + lane];
    float d = h0 * a_dst[lane] + h1 * a_dst[32 + lane];
    #pragma unroll
    for (int off = 16; off > 0; off >>= 1) {
        s += __shfl_down(s, off, 32);
        d += __shfl_down(d, off, 32);
    }
    if (lane == 0) {
        ls[node] = s;
        ld_[node] = d;
        mx[node] = -INFINITY;
        denom[node] = 0.0f;
    }
    outp[base + lane]      = bias[lane];
    outp[base + 32 + lane] = bias[32 + lane];
}

// Float atomic max via ordered-int trick (lowers to global_atomic_max/min).
__device__ __forceinline__ void atomicMaxF(float* addr, float v) {
    if (v >= 0.0f) atomicMax((int*)addr, __float_as_int(v));
    else           atomicMin((unsigned int*)addr, __float_as_uint(v));
}

// Edge phase 1: e = leakyrelu(ls[src] + ld[dst]); segment max over dst.
// Edges [0,E) from the graph, [E, E+N) are self loops.
__global__ __launch_bounds__(256) void edge_max_kernel(
    const long long* __restrict__ src, const long long* __restrict__ dst,
    const float* __restrict__ ls, const float* __restrict__ ld_,
    float* __restrict__ ebuf, float* __restrict__ mx, int E, int N)
{
    int j = blockIdx.x * 256 + threadIdx.x;
    int Et = E + N;
    if (j >= Et) return;
    int s, d;
    if (j < E) { s = (int)src[j]; d = (int)dst[j]; }
    else       { s = j - E;       d = s; }
    float e = ls[s] + ld_[d];
    e = (e > 0.0f) ? e : NEG_SLOPE * e;
    ebuf[j] = e;
    atomicMaxF(&mx[d], e);
}

// Edge phase 2: ex = exp(e - m[dst]); denom[dst] += ex.
__global__ __launch_bounds__(256) void edge_exp_kernel(
    const long long* __restrict__ dst_,
    float* __restrict__ ebuf, const float* __restrict__ mx,
    float* __restrict__ denom, int E, int N)
{
    int j = blockIdx.x * 256 + threadIdx.x;
    int Et = E + N;
    if (j >= Et) return;
    int d = (j < E) ? (int)dst_[j] : (j - E);
    float ex = expf(ebuf[j] - mx[d]);
    ebuf[j] = ex;
    atomicAdd(&denom[d], ex);
}

// Edge phase 3: out[dst] += (ex / (denom[dst]+EPS)) * h[src].  Wave per edge.
__global__ __launch_bounds__(256) void edge_scatter_kernel(
    const long long* __restrict__ src, const long long* __restrict__ dst,
    const float* __restrict__ ebuf, const float* __restrict__ denom,
    const float* __restrict__ h, float* __restrict__ outp, int E, int N)
{
    int j = blockIdx.x * 8 + (threadIdx.x >> 5);
    int lane = threadIdx.x & 31;
    int Et = E + N;
    if (j >= Et) return;
    int s, d;
    if (j < E) { s = (int)src[j]; d = (int)dst[j]; }
    else       { s = j - E;       d = s; }
    float alpha = ebuf[j] / (denom[d] + EPS_F);
    size_t sb = (size_t)s * 64, db = (size_t)d * 64;
    atomicAdd(&outp[db + lane],      alpha * h[sb + lane]);
    atomicAdd(&outp[db + 32 + lane], alpha * h[sb + 32 + lane]);
}

// ---------------------------------------------------------------------------
extern "C" void kernel_launch(void* const* d_in, const int* in_sizes, int n_in,
                              void* d_out, int out_size, void* d_ws, size_t ws_size,
                              hipStream_t stream)
{
    const float*     x     = (const float*)d_in[0];
    const long long* edges = (const long long*)d_in[1];   // int64 in reference
    const float*     W     = (const float*)d_in[2];
    const float*     a_src = (const float*)d_in[3];
    const float*     a_dst = (const float*)d_in[4];
    const float*     gbias = (const float*)d_in[5];
    const float*     w1    = (const float*)d_in[6];
    const float*     b1    = (const float*)d_in[7];
    const float*     w2    = (const float*)d_in[8];
    const float*     b2    = (const float*)d_in[9];

    const int N  = in_sizes[0] / 64;
    const int L  = in_sizes[2] / (64 * 64);
    const int E  = (int)((long long)in_sizes[1] / (2LL * L));
    const int Et = E + N;

    // Workspace (~82 MB): never materialize the N x (L+1)*64 concat; each
    // layer's output is folded into acc += out_l @ w1_block immediately.
    float* ws    = (float*)d_ws;
    float* h     = ws;  ws += (size_t)N * 64;
    float* outp  = ws;  ws += (size_t)N * 64;
    float* acc   = ws;  ws += (size_t)N * 64;
    float* ls    = ws;  ws += N;
    float* ldl   = ws;  ws += N;
    float* mx    = ws;  ws += N;
    float* denom = ws;  ws += N;
    float* ebuf  = ws;  ws += Et;

    const int gemmGrid  = (N + 31) / 32;
    const int nodeGrid  = (N + 7) / 8;
    const int edgeGridT = (Et + 255) / 256;
    const int edgeGridW = (Et + 7) / 8;

    // acc = x @ w1[0:64,:] + b1  (first block of the fused concat-GEMM)
    gemm64_kernel<false, false, true, false><<<gemmGrid, 256, 0, stream>>>(
        x, w1, nullptr, b1, acc, N);

    for (int l = 0; l < L; ++l) {
        const long long* esrc = edges + (size_t)l * 2 * E;
        const long long* edst = esrc + E;

        // h = x @ W_l
        gemm64_kernel<false, false, false, false><<<gemmGrid, 256, 0, stream>>>(
            x, W + (size_t)l * 4096, nullptr, nullptr, h, N);

        node_prep_kernel<<<nodeGrid, 256, 0, stream>>>(
            h, a_src + (size_t)l * 64, a_dst + (size_t)l * 64,
            gbias + (size_t)l * 64, ls, ldl, mx, denom, outp, N);

        edge_max_kernel<<<edgeGridT, 256, 0, stream>>>(
            esrc, edst, ls, ldl, ebuf, mx, E, N);

        edge_exp_kernel<<<edgeGridT, 256, 0, stream>>>(
            edst, ebuf, mx, denom, E, N);

        edge_scatter_kernel<<<edgeGridW, 256, 0, stream>>>(
            esrc, edst, ebuf, denom, h, outp, E, N);

        // acc += out_l @ w1[(l+1)*64 : (l+2)*64, :]
        gemm64_kernel<false, true, false, false><<<gemmGrid, 256, 0, stream>>>(
            outp, w1 + (size_t)(l + 1) * 4096, acc, nullptr, acc, N);
    }

    // out = tanh( tanh(acc) @ w2 + b2 )
    gemm64_kernel<true, false, true, true><<<gemmGrid, 256, 0, stream>>>(
        acc, w2, nullptr, b2, (float*)d_out, N);
}